// CenterLoss_25305947308120
// MI455X (gfx1250) — compile-verified
//
#include <hip/hip_runtime.h>

// CenterLoss on gfx1250 (MI455X).
// Memory-bound: 128 MB feature stream / 23.3 TB/s ~ 5.6 us floor.
// Per-sample ||f - c||^2 computed as diag(diff @ diff^T) via V_WMMA_F32_16X16X4_F32
// (f32 WMMA => bitwise-compatible precision with the f32 reference accumulate).

typedef __attribute__((ext_vector_type(2))) float v2f;
typedef __attribute__((ext_vector_type(8))) float v8f;

#define FEAT_DIM    512
#define NUM_CLASSES 1000
#define BATCH       65536

// ---------------- init: zero per-class sums + counts in workspace -------------
__global__ void cl_init(float* __restrict__ ws) {
  int i = blockIdx.x * blockDim.x + threadIdx.x;
  if (i < 2 * NUM_CLASSES) ws[i] = 0.0f;
}

// ---------------- main: one wave = 16 samples, WMMA diag reduction ------------
__global__ __launch_bounds__(256) void cl_main(const float* __restrict__ feat,
                                               const float* __restrict__ cent,
                                               const int*   __restrict__ labels,
                                               float* __restrict__ sums,
                                               float* __restrict__ counts) {
  const int lane   = threadIdx.x & 31;
  const int waveId = blockIdx.x * (blockDim.x >> 5) + (threadIdx.x >> 5);
  const int base   = waveId * 16;        // 16-sample tile per wave (65536/16 = 4096 waves)
  const int m      = lane & 15;          // tile row (sample) this lane feeds
  const int s      = base + m;
  const int half   = lane >> 4;          // lanes 0-15: K+{0,1}; lanes 16-31: K+{2,3}

  const int lab = labels[s];             // lanes 16-31 duplicate lanes 0-15 (same line)

  const float* __restrict__ frow = feat + (size_t)s   * FEAT_DIM;
  const float* __restrict__ crow = cent + (size_t)lab * FEAT_DIM;

  // Two accumulators to break the WMMA D->C dependence chain.
  v8f acc0 = {};
  v8f acc1 = {};

#pragma unroll 4
  for (int k = 0; k < FEAT_DIM; k += 8) {
    const int k0 = k + half * 2;         // K-chunk [k, k+4)
    const int k1 = k + 4 + half * 2;     // K-chunk [k+4, k+8)

    if ((k & 24) == 0)                   // one speculative line-ahead prefetch per 32 floats
      __builtin_prefetch(frow + k + 32, 0, 1);

    v2f f0 = *(const v2f*)(frow + k0);
    v2f c0 = *(const v2f*)(crow + k0);
    v2f f1 = *(const v2f*)(frow + k1);
    v2f c1 = *(const v2f*)(crow + k1);
    v2f d0 = f0 - c0;
    v2f d1 = f1 - c1;

    // A (16x4) and B (4x16 = A^T) have identical per-lane register images for
    // this diff tile, so the same v2f feeds both operands: acc += A*A^T.
    acc0 = __builtin_amdgcn_wmma_f32_16x16x4_f32(false, d0, false, d0,
                                                 (short)0, acc0, false, false);
    acc1 = __builtin_amdgcn_wmma_f32_16x16x4_f32(false, d1, false, d1,
                                                 (short)0, acc1, false, false);
  }
  v8f acc = acc0 + acc1;                 // layouts match -> elementwise add is valid

  // Diagonal extraction: D[i][i] at (VGPR i, lane i) for i<8,
  //                      (VGPR i-8, lane i+16) for i>=8.
  float d = 0.0f;
#pragma unroll
  for (int r = 0; r < 8; ++r) {
    float v  = acc[r];
    float hi = __shfl(v, lane + 16, 32); // lane r+8 reads lane r+24 -> D[r+8][r+8]
    if (lane == r)     d = v;
    if (lane == r + 8) d = hi;
  }

  if (lane < 16) {                       // lanes 0-15 own samples base..base+15
    atomicAdd(&sums[lab], d);
    atomicAdd(&counts[lab], 1.0f);
  }
}

// ---------------- finalize: sum_c sums/(counts*512) / BATCH -------------------
__global__ void cl_final(const float* __restrict__ sums,
                         const float* __restrict__ counts,
                         float* __restrict__ out) {
  __shared__ float red[256];
  float local = 0.0f;
  for (int c = threadIdx.x; c < NUM_CLASSES; c += 256) {
    float cnt = counts[c];
    if (cnt > 0.0f) local += sums[c] / (cnt * (float)FEAT_DIM);
  }
  red[threadIdx.x] = local;
  __syncthreads();
  for (int off = 128; off > 0; off >>= 1) {
    if (threadIdx.x < off) red[threadIdx.x] += red[threadIdx.x + off];
    __syncthreads();
  }
  if (threadIdx.x == 0) out[0] = red[0] / (float)BATCH;
}

extern "C" void kernel_launch(void* const* d_in, const int* in_sizes, int n_in,
                              void* d_out, int out_size, void* d_ws, size_t ws_size,
                              hipStream_t stream) {
  const float* feat   = (const float*)d_in[0];   // [65536, 512] f32
  const float* cent   = (const float*)d_in[1];   // [1000, 512]  f32
  const int*   labels = (const int*)  d_in[2];   // [65536]      int32

  float* sums   = (float*)d_ws;                  // [1000]
  float* counts = sums + NUM_CLASSES;            // [1000]

  cl_init<<<(2 * NUM_CLASSES + 255) / 256, 256, 0, stream>>>(sums);

  // 4096 waves of 16 samples; 8 waves (256 threads) per block -> 512 blocks.
  cl_main<<<BATCH / (16 * 8), 256, 0, stream>>>(feat, cent, labels, sums, counts);

  cl_final<<<1, 256, 0, stream>>>(sums, counts, (float*)d_out);
}